// GAPCutLoss_60988535603742
// MI455X (gfx1250) — compile-verified
//
#include <hip/hip_runtime.h>
#include <stdint.h>

#define NN 10000
#define KK 16
#define MEAN_NODES 625.0f

typedef __attribute__((ext_vector_type(2))) float v2f;
typedef __attribute__((ext_vector_type(4))) float v4f;
typedef __attribute__((ext_vector_type(8))) float v8f;
typedef __attribute__((ext_vector_type(4))) unsigned int u32x4;
typedef __attribute__((ext_vector_type(4))) int i32x4;
typedef __attribute__((ext_vector_type(8))) int i32x8;

constexpr int TJ = 256;              // columns of A staged per tile
constexpr int TILES_PER_CHUNK = 5;   // 8 chunks * 5 tiles * 256 = 10240 >= 10000
constexpr int CHUNKS = 8;
constexpr int ROWB = 625;            // 625 * 16 = 10000 rows
constexpr int LP = 260;              // LDS row stride in floats = 256 + TDM pad of 4 DWORDs
                                     // (stride mod 64 banks = 4 -> conflict-free b64 frag reads)

#define WS_GAMMA 0
#define WS_EPART 16
#define WS_PART  32
#define NPART (ROWB * CHUNKS)

// ---------------- Kernel 1: gamma_k = sum_i Y[i,k]*deg[i]; colsum_k; error_partition ----
__global__ __launch_bounds__(1024)
void gap_stats(const float* __restrict__ Y, const float* __restrict__ deg,
               float* __restrict__ ws) {
  const int tid = threadIdx.x;
  float g[KK], cs[KK];
#pragma unroll
  for (int k = 0; k < KK; ++k) { g[k] = 0.f; cs[k] = 0.f; }
  for (int i = tid; i < NN; i += 1024) {
    const float d = deg[i];
    const float* yr = Y + (long)i * KK;
#pragma unroll
    for (int k = 0; k < KK; ++k) { const float y = yr[k]; g[k] += y * d; cs[k] += y; }
  }
#pragma unroll
  for (int k = 0; k < KK; ++k) {
#pragma unroll
    for (int off = 16; off >= 1; off >>= 1) {
      g[k]  += __shfl_xor(g[k],  off, 32);
      cs[k] += __shfl_xor(cs[k], off, 32);
    }
  }
  __shared__ float sg[32][KK];
  __shared__ float sc[32][KK];
  const int wave = tid >> 5, lane = tid & 31;
  if (lane == 0) {
#pragma unroll
    for (int k = 0; k < KK; ++k) { sg[wave][k] = g[k]; sc[wave][k] = cs[k]; }
  }
  __syncthreads();
  if (tid == 0) {
    float ep = 0.f;
    for (int k = 0; k < KK; ++k) {
      float G = 0.f, C = 0.f;
      for (int w = 0; w < 32; ++w) { G += sg[w][k]; C += sc[w][k]; }
      ws[WS_GAMMA + k] = G;
      const float dd = C - MEAN_NODES;
      ep += dd * dd;
    }
    ws[WS_EPART] = ep;
  }
}

// ---- TDM: DMA one 16 x 256 f32 tile (row stride NN) into LDS with 4-DWORD row padding --
// D# per CDNA5 ISA. dim0 = remaining valid columns: TDM
// zero-fills reads beyond tensor_dim0, which natively masks the ragged right edge.
__device__ __forceinline__ void tdm_load_tile(const float* gp, unsigned lds_off,
                                              unsigned dim0) {
  const unsigned long long ga = (unsigned long long)(uintptr_t)gp;
  u32x4 g0 = {
      1u,                                                // count=1, is_restore=0
      lds_off,                                           // lds_addr (bytes)
      (unsigned)(ga & 0xFFFFFFFFu),                      // global_addr[31:0]
      (unsigned)((ga >> 32) & 0x01FFFFFFu) | (2u << 30)  // global_addr[56:32] | type=2
  };
  i32x8 g1 = {
      (int)((2u << 16) | (1u << 20) | (7u << 22) | (3u << 25)),
      //    data_size=4B | pad_enable | pad_interval: 256 DW | pad_amount: 4 DW
      (int)((dim0 & 0xFFFFu) << 16),                     // tensor_dim0[15:0] @ bits[63:48]
      (int)(((dim0 >> 16) & 0xFFFFu) | (16u << 16)),     // dim0[31:16] | tensor_dim1=16
      (int)(256u << 16),                                 // tile_dim0 = 256
      16,                                                // tile_dim1 = 16, tile_dim2 = 0
      NN,                                                // tensor_dim0_stride[31:0]
      0,                                                 // stride hi | tensor_dim1_stride lo
      0
  };
  const i32x4 z4 = {0, 0, 0, 0};                         // groups 2/3 unused (2D tile)
  const i32x8 z8 = {0, 0, 0, 0, 0, 0, 0, 0};
  __builtin_amdgcn_tensor_load_to_lds(g0, g1, z4, z4, z8, 0);
}

// ---- per-tile A row sums from the staged LDS tile (wave w owns rows 2w, 2w+1) ----------
__device__ __forceinline__ void rowsum_tile(const float* __restrict__ buf,
                                            int wave, int lane, float& rs0, float& rs1) {
#pragma unroll
  for (int e = 0; e < 2; ++e) {
    const int r = 2 * wave + e;
    float s = 0.f;
#pragma unroll
    for (int h = 0; h < 2; ++h) {
      v4f v = *(const v4f*)(&buf[r * LP + h * 128 + lane * 4]);   // ds_load_b128
      s += (v.x + v.y) + (v.z + v.w);
    }
    if (e == 0) rs0 += s; else rs1 += s;
  }
}

// ---- 8x WMMA 16x16x4 f32 over one staged tile; wave owns local cols [32w, 32w+32) -----
template <bool FULL>
__device__ __forceinline__ void compute_tile(const float* __restrict__ Y,
                                             const float* __restrict__ buf,
                                             int wave, int l15, int lhi, long j0,
                                             v8f& c) {
#pragma unroll
  for (int t4 = 0; t4 < 8; ++t4) {
    const int cbase = 32 * wave + 4 * t4 + 2 * lhi;           // K-pair start
    v2f a = *(const v2f*)(&buf[l15 * LP + cbase]);            // ds_load_b64, conflict-free
    const long jrow = j0 + cbase;
    v2f b;
    if (FULL) {
      b.x = Y[jrow * KK + l15];
      b.y = Y[(jrow + 1) * KK + l15];
    } else {
      b.x = (jrow     < NN) ? Y[jrow * KK + l15]       : 0.f;
      b.y = (jrow + 1 < NN) ? Y[(jrow + 1) * KK + l15] : 0.f;
    }
    c = __builtin_amdgcn_wmma_f32_16x16x4_f32(false, a, false, b, (short)0, c,
                                              false, false);
  }
}

// ---- TDM-double-buffered streaming loop over one chunk ---------------------------------
template <bool FULL>
__device__ __forceinline__ void run_chunk(const float* __restrict__ A,
                                          const float* __restrict__ Y,
                                          const float* b0, const float* b1,
                                          unsigned lds0, unsigned lds1,
                                          int wave, int lane, int l15, int lhi,
                                          int row0, long jbase,
                                          v8f& c, float& rs0, float& rs1) {
  const bool issuer = (threadIdx.x < 32);                 // wave 0 drives the TDM
  const float* arow0 = A + (long)row0 * NN;
  if (issuer) {
    tdm_load_tile(arow0 + jbase, lds0, (unsigned)(NN - jbase));
    __builtin_amdgcn_s_wait_tensorcnt(0);
  }
  __syncthreads();
#pragma unroll
  for (int t = 0; t < TILES_PER_CHUNK; ++t) {
    const float* cur = (t & 1) ? b1 : b0;
    const unsigned nxtoff = (t & 1) ? lds0 : lds1;
    const long j0 = jbase + (long)t * TJ;
    if ((t + 1 < TILES_PER_CHUNK) && issuer)              // DMA next tile, ASYNC w/ compute
      tdm_load_tile(arow0 + (j0 + TJ), nxtoff, (unsigned)(NN - (j0 + TJ)));
    rowsum_tile(cur, wave, lane, rs0, rs1);
    compute_tile<FULL>(Y, cur, wave, l15, lhi, j0, c);
    if (issuer) __builtin_amdgcn_s_wait_tensorcnt(0);     // next tile landed
    __syncthreads();                                      // flip buffers
  }
}

// ---------------- Kernel 2: stream A via TDM, B = A@Y via V_WMMA_F32_16X16X4_F32 --------
__global__ __launch_bounds__(256)
void gap_main(const float* __restrict__ Y, const float* __restrict__ A,
              const float* __restrict__ ws, float* __restrict__ partials) {
  __shared__ __align__(16) float atile0[16 * LP];
  __shared__ __align__(16) float atile1[16 * LP];
  __shared__ float btile[16 * KK + 16];   // 16x16 B-tile sums + 16 rowsums

  const int tid  = threadIdx.x;
  const int wave = tid >> 5;
  const int lane = tid & 31;
  const int l15  = lane & 15;
  const int lhi  = lane >> 4;
  const int row0 = blockIdx.x * 16;
  const long jbase = (long)blockIdx.y * (TILES_PER_CHUNK * TJ);

  const unsigned lds0 = (unsigned)(uintptr_t)atile0;      // LDS aperture: low 32 bits = offset
  const unsigned lds1 = (unsigned)(uintptr_t)atile1;

  v8f c = {};
  float rs0 = 0.f, rs1 = 0.f;

  const bool full = (jbase + (long)TILES_PER_CHUNK * TJ) <= NN;  // uniform per block
  if (full)
    run_chunk<true >(A, Y, atile0, atile1, lds0, lds1, wave, lane, l15, lhi, row0, jbase,
                     c, rs0, rs1);
  else
    run_chunk<false>(A, Y, atile0, atile1, lds0, lds1, wave, lane, l15, lhi, row0, jbase,
                     c, rs0, rs1);

  // ---- reduce per-lane A row sums across the wave ----
#pragma unroll
  for (int off = 16; off >= 1; off >>= 1) {
    rs0 += __shfl_xor(rs0, off, 32);
    rs1 += __shfl_xor(rs1, off, 32);
  }

  // ---- cross-wave accumulation of the 16x16 tile + rowsums in LDS ----
  for (int i = tid; i < 16 * KK + 16; i += 256) btile[i] = 0.f;
  __syncthreads();
  if (lane == 0) {                       // rows are exclusive per wave: plain stores
    btile[16 * KK + 2 * wave + 0] = rs0;
    btile[16 * KK + 2 * wave + 1] = rs1;
  }
  {
    const int mbase = 8 * lhi;           // C/D layout: VGPR v -> M = v + 8*(lane>=16)
#pragma unroll
    for (int v = 0; v < 8; ++v) atomicAdd(&btile[(mbase + v) * KK + l15], c[v]);
  }
  __syncthreads();

  // ---- fold in gamma: partial = sum_i [ s_i * rowA_i - dot(Yn_i, B_i) ] ----
  if (tid == 0) {
    float acc = 0.f;
    for (int r = 0; r < 16; ++r) {
      const float* yr = Y + (long)(row0 + r) * KK;
      float s = 0.f, dot = 0.f;
      for (int k = 0; k < KK; ++k) {
        const float yn = yr[k] / ws[WS_GAMMA + k];
        s   += yn;
        dot += yn * btile[r * KK + k];
      }
      acc += s * btile[16 * KK + r] - dot;
    }
    partials[(long)blockIdx.y * gridDim.x + blockIdx.x] = acc;
  }
}

// ---------------- Kernel 3: reduce 5000 block partials + error_partition -> scalar -----
__global__ __launch_bounds__(256)
void gap_finish(const float* __restrict__ ws, float* __restrict__ out) {
  const int tid = threadIdx.x;
  float acc = 0.f;
  for (int i = tid; i < NPART; i += 256) acc += ws[WS_PART + i];
#pragma unroll
  for (int off = 16; off >= 1; off >>= 1) acc += __shfl_xor(acc, off, 32);
  __shared__ float sw[8];
  if ((tid & 31) == 0) sw[tid >> 5] = acc;
  __syncthreads();
  if (tid == 0) {
    float tot = ws[WS_EPART];
    for (int w = 0; w < 8; ++w) tot += sw[w];
    out[0] = tot;
  }
}

extern "C" void kernel_launch(void* const* d_in, const int* in_sizes, int n_in,
                              void* d_out, int out_size, void* d_ws, size_t ws_size,
                              hipStream_t stream) {
  (void)in_sizes; (void)n_in; (void)out_size; (void)ws_size;
  const float* Y   = (const float*)d_in[0];
  const float* A   = (const float*)d_in[1];
  const float* deg = (const float*)d_in[2];
  float* ws  = (float*)d_ws;
  float* out = (float*)d_out;

  gap_stats<<<1, 1024, 0, stream>>>(Y, deg, ws);
  gap_main<<<dim3(ROWB, CHUNKS), 256, 0, stream>>>(Y, A, ws, ws + WS_PART);
  gap_finish<<<1, 256, 0, stream>>>(ws, out);
}